// UserGNNEncoder_48816598286984
// MI455X (gfx1250) — compile-verified
//
#include <hip/hip_runtime.h>

typedef float v2f __attribute__((ext_vector_type(2)));
typedef float v8f __attribute__((ext_vector_type(8)));

// ---------------------------------------------------------------------------
// Scatter-add: one wave32 per edge. Gathers x[src] (float4 per lane, coalesced
// 512B per wave) and atomically accumulates into agg[dst]; lane 0 counts the
// edge for the mean denominator. Atomics execute in L2 (accumulators fit in
// the 192MB L2), so this runs at memory-system rate.
// ---------------------------------------------------------------------------
__global__ void scatter_add_kernel(const long long* __restrict__ ei,
                                   const float* __restrict__ x,
                                   float* __restrict__ agg,
                                   float* __restrict__ cnt,
                                   int E, int D) {
  const int gid  = blockIdx.x * blockDim.x + threadIdx.x;
  const int e    = gid >> 5;
  const int lane = gid & 31;
  if (e >= E) return;
  const long long src = ei[e];
  const long long dst = ei[(size_t)E + e];
  const float* __restrict__ xr = x + (size_t)src * D;
  float* __restrict__ ar = agg + (size_t)dst * D;
  for (int c = lane * 4; c < D; c += 32 * 4) {
    const float4 v = *reinterpret_cast<const float4*>(xr + c);
    atomicAdd(ar + c + 0, v.x);
    atomicAdd(ar + c + 1, v.y);
    atomicAdd(ar + c + 2, v.z);
    atomicAdd(ar + c + 3, v.w);
  }
  if (cnt != nullptr && lane == 0) atomicAdd(cnt + dst, 1.0f);
}

// ---------------------------------------------------------------------------
// Fused SAGE linear:  out = act( (A1/max(cnt,1)) @ W1  +  A2 @ W2  + bias )
//   - 256 threads = 8 waves; workgroup tile 128(M) x 64(N); each wave owns a
//     16x64 strip = four 16x16 f32 WMMA accumulators.
//   - Weights stream through DOUBLE-BUFFERED LDS via gfx1250 async copies
//     (global_load_async_to_lds_b128, ASYNCcnt): block kb+1 loads while
//     block kb computes; s_wait_asynccnt 0 + barrier at the block boundary.
//   - Requires K1, K2 to be multiples of TK (holds here: K in {128,256}).
//   - WMMA under uniform control flow only (EXEC all-1s); M-edge handled by
//     clamped addresses + zero masks, stores predicated per element.
// ---------------------------------------------------------------------------
#define TM 128
#define TN 64
#define TK 64

// Issue async copies of W[kb:kb+TK, n0:n0+TN] into the LDS buffer.
// 4096 floats = 1024 float4 chunks; 256 threads -> 4 chunks each, coalesced in N.
__device__ __forceinline__ void stage_w_async(float* sdst,
                                              const float* __restrict__ W,
                                              int kb, int N, int n0, int tid) {
#pragma unroll
  for (int c = 0; c < (TK * TN) / (4 * 256); ++c) {
    const int p  = c * 256 + tid;       // float4 index in tile
    const int kk = p >> 4;              // TN/4 = 16 float4 per row
    const int n4 = (p & 15) * 4;
    const float* g = W + (size_t)(kb + kk) * N + (n0 + n4);
    const unsigned lds = (unsigned)(size_t)(sdst + kk * TN + n4);  // LDS offset
    asm volatile("global_load_async_to_lds_b128 %0, %1, off"
                 :: "v"(lds), "v"(g) : "memory");
  }
}

__device__ __forceinline__ void wait_async_all() {
  asm volatile("s_wait_asynccnt 0x0" ::: "memory");
}

__global__ __launch_bounds__(256)
void fused_sage_gemm(const float* __restrict__ A1, const float* __restrict__ cnt,
                     const float* __restrict__ W1,
                     const float* __restrict__ A2, const float* __restrict__ W2,
                     const float* __restrict__ bias,
                     float* __restrict__ out,
                     int M, int K1, int K2, int N, int do_relu) {
  __shared__ float sW[2][TK * TN];      // 2 x 16KB double buffer

  const int tid   = threadIdx.x;
  const int wv    = tid >> 5;
  const int lane  = tid & 31;
  const int lh    = lane >> 4;   // 0: lanes 0-15 (K pair 0,1) ; 1: lanes 16-31 (K pair 2,3)
  const int lm    = lane & 15;
  const int mbase = blockIdx.x * TM + wv * 16;
  const int n0    = blockIdx.y * TN;

  v8f acc[4] = {};

  // A-matrix 16x4 fragment: lane (lh,lm) supplies row (mbase+lm), K cols {2*lh, 2*lh+1}.
  const int   rowA   = mbase + lm;
  const int   rowAc  = rowA < M ? rowA : (M - 1);
  const float mvalid = (rowA < M) ? 1.0f : 0.0f;

  for (int phase = 0; phase < 2; ++phase) {
    const float* __restrict__ A = phase ? A2 : A1;
    const float* __restrict__ W = phase ? W2 : W1;
    const int K = phase ? K2 : K1;
    if (K == 0 || A == nullptr) continue;   // uniform branch: EXEC stays all-1s

    float scale = mvalid;
    if (phase == 0 && cnt != nullptr) {
      scale = mvalid / fmaxf(cnt[rowAc], 1.0f);   // fold mean into A fragment
    }
    const float* __restrict__ Arow = A + (size_t)rowAc * K;

    // Prologue: async-stage first K block, wait, publish.
    int nbuf = 0;
    stage_w_async(sW[0], W, 0, N, n0, tid);
    wait_async_all();
    __syncthreads();

    for (int kb = 0; kb < K; kb += TK) {
      // Overlap: kick off next block's async copies before computing this one.
      if (kb + TK < K) stage_w_async(sW[nbuf ^ 1], W, kb + TK, N, n0, tid);

      const float* __restrict__ sb = sW[nbuf];
#pragma unroll 4
      for (int kk = 0; kk < TK; kk += 4) {
        // A fragment: aligned float2 (K multiple of 4).
        const float2 av = *reinterpret_cast<const float2*>(Arow + kb + kk + lh * 2);
        v2f a;
        a.x = av.x * scale;
        a.y = av.y * scale;
#pragma unroll
        for (int t = 0; t < 4; ++t) {
          // B 4x16 fragment from LDS: VGPR0 = K {kk+2*lh}, VGPR1 = K {kk+2*lh+1}, N = lm.
          v2f b;
          b.x = sb[(kk + lh * 2 + 0) * TN + t * 16 + lm];
          b.y = sb[(kk + lh * 2 + 1) * TN + t * 16 + lm];
          acc[t] = __builtin_amdgcn_wmma_f32_16x16x4_f32(
              false, a, false, b, (short)0, acc[t], false, false);
        }
      }

      // Block boundary: my async writes to the other buffer are done, and my
      // LDS reads of this buffer are consumed; one barrier publishes both.
      wait_async_all();
      __syncthreads();
      nbuf ^= 1;
    }
  }

  // Epilogue: bias + optional ReLU + predicated stores.
  // C layout: VGPR r -> M = mbase + r + 8*lh, N = n0 + 16*t + lm.
#pragma unroll
  for (int t = 0; t < 4; ++t) {
    const int   n  = n0 + t * 16 + lm;
    const float bv = (bias != nullptr) ? bias[n] : 0.0f;
#pragma unroll
    for (int r = 0; r < 8; ++r) {
      const int m = mbase + r + lh * 8;
      if (m < M) {
        float v = acc[t][r] + bv;
        if (do_relu) v = fmaxf(v, 0.0f);
        out[(size_t)m * N + n] = v;
      }
    }
  }
}

// ---------------------------------------------------------------------------
// Orchestration
// ---------------------------------------------------------------------------
extern "C" void kernel_launch(void* const* d_in, const int* in_sizes, int n_in,
                              void* d_out, int out_size, void* d_ws, size_t ws_size,
                              hipStream_t stream) {
  const float*     x_movie  = (const float*)d_in[0];
  const float*     x_user   = (const float*)d_in[1];
  const long long* ei_sims  = (const long long*)d_in[2];
  const long long* ei_rates = (const long long*)d_in[3];
  const float* c1_Wl = (const float*)d_in[4];
  const float* c1_bl = (const float*)d_in[5];
  const float* c1_Wr = (const float*)d_in[6];
  const float* c2_Wl = (const float*)d_in[7];
  const float* c2_bl = (const float*)d_in[8];
  const float* c2_Wr = (const float*)d_in[9];
  const float* c3_Wl = (const float*)d_in[10];
  const float* c3_bl = (const float*)d_in[11];
  const float* c3_Wr = (const float*)d_in[12];
  const float* l1_W  = (const float*)d_in[13];
  const float* l1_b  = (const float*)d_in[14];
  const float* l2_W  = (const float*)d_in[15];
  const float* l2_b  = (const float*)d_in[16];
  const float* l3_W  = (const float*)d_in[17];
  const float* l3_b  = (const float*)d_in[18];

  const int D = 128, H = 256, O = 64;
  const int NM = in_sizes[0] / D;     // 50000
  const int NU = in_sizes[1] / D;     // 100000
  const int E  = in_sizes[2] / 2;     // 600000

  // Workspace layout with lifetime-based overlays (~385MB peak).
  char* ws = (char*)d_ws;
  const size_t fNM_D = (size_t)NM * D * sizeof(float);   // 25.6 MB
  const size_t fNM_H = (size_t)NM * H * sizeof(float);   // 51.2 MB
  const size_t fNU_D = (size_t)NU * D * sizeof(float);   // 51.2 MB
  const size_t fNU_H = (size_t)NU * H * sizeof(float);   // 102.4 MB

  float* agg1 = (float*)(ws);                              // dead after gemm1
  float* h    = (float*)(ws + fNM_D);                      // dead after gemm2
  float* h2   = (float*)(ws + fNM_D + fNM_H);              // dead after scatter3
  float* u3   = (float*)(ws);                              // overlays agg1/h/h2 (all dead)
  const size_t r2 = fNM_D + 2 * fNM_H;
  float* agg2 = (float*)(ws + r2);                         // dead after gemm3
  float* u    = (float*)(ws + r2 + fNU_D);                 // dead after gemm4
  float* agg3 = (float*)(ws + r2);                         // overlays agg2+u (dead)
  float* u2   = (float*)(ws + r2 + fNU_D + fNU_H);
  float* cnt_m = (float*)(ws + r2 + fNU_D + 2 * fNU_H);
  float* cnt_u = cnt_m + NM;
  (void)ws_size; (void)n_in; (void)out_size;

  const dim3 blk(256);
  const dim3 gM((NM + TM - 1) / TM, H / TN);
  const dim3 gU((NU + TM - 1) / TM, H / TN);
  const dim3 gO((NU + TM - 1) / TM, O / TN);
  const int  sgrid = (E * 32 + 255) / 256;    // one wave32 per edge

  // ---- layer 1: movie-movie SAGE + linear ----
  hipMemsetAsync(agg1, 0, fNM_D, stream);
  hipMemsetAsync(cnt_m, 0, (size_t)NM * sizeof(float), stream);
  scatter_add_kernel<<<sgrid, 256, 0, stream>>>(ei_sims, x_movie, agg1, cnt_m, E, D);
  fused_sage_gemm<<<gM, blk, 0, stream>>>(agg1, cnt_m, c1_Wl, x_movie, c1_Wr,
                                          c1_bl, h, NM, D, D, H, 1);
  fused_sage_gemm<<<gM, blk, 0, stream>>>(h, nullptr, l1_W, nullptr, nullptr,
                                          l1_b, h2, NM, H, 0, H, 1);

  // ---- layer 2: movie-user SAGE + linear ----
  hipMemsetAsync(agg2, 0, fNU_D, stream);
  hipMemsetAsync(cnt_u, 0, (size_t)NU * sizeof(float), stream);
  scatter_add_kernel<<<sgrid, 256, 0, stream>>>(ei_rates, x_movie, agg2, cnt_u, E, D);
  fused_sage_gemm<<<gU, blk, 0, stream>>>(agg2, cnt_u, c2_Wl, x_user, c2_Wr,
                                          c2_bl, u, NU, D, D, H, 1);
  fused_sage_gemm<<<gU, blk, 0, stream>>>(u, nullptr, l2_W, nullptr, nullptr,
                                          l2_b, u2, NU, H, 0, H, 1);

  // ---- layer 3: h2 -> users SAGE (reuse cnt_u: same edge list) + head ----
  hipMemsetAsync(agg3, 0, fNU_H, stream);
  scatter_add_kernel<<<sgrid, 256, 0, stream>>>(ei_rates, h2, agg3, nullptr, E, H);
  fused_sage_gemm<<<gU, blk, 0, stream>>>(agg3, cnt_u, c3_Wl, u2, c3_Wr,
                                          c3_bl, u3, NU, H, H, H, 1);
  fused_sage_gemm<<<gO, blk, 0, stream>>>(u3, nullptr, l3_W, nullptr, nullptr,
                                          l3_b, (float*)d_out, NU, H, 0, O, 0);
}